// GNN_45208825757774
// MI455X (gfx1250) — compile-verified
//
#include <hip/hip_runtime.h>
#include <hip/hip_bf16.h>

// GCN 2-layer forward for MI455X (gfx1250), wave32.
// GEMMs use native fp32 WMMA: v_wmma_f32_16x16x4_f32 (no precision loss vs fp32 ref).
// Aggregation is edge-parallel float4 gather + global_atomic_add_f32 scatter (L2-resident).
// d_out doubles as the second ping-pong buffer -> ws footprint ~55 MB.

#define GCN_N_NODES 50000
#define GCN_N_EDGES 800000
#define GCN_IN_CH   128
#define GCN_HID     256

typedef __attribute__((ext_vector_type(2))) float v2f;
typedef __attribute__((ext_vector_type(8))) float v8f;

// ---------------------------------------------------------------- fill
__global__ void k_fill_f4(float4* __restrict__ p, float v, int n4) {
    int i = blockIdx.x * blockDim.x + threadIdx.x;
    if (i < n4) p[i] = make_float4(v, v, v, v);
}

// ---------------------------------------------------------------- degree count (self-loop pre-seeded to 1.0)
__global__ void k_deg_count(const int* __restrict__ dst, float* __restrict__ deg, int E) {
    int e = blockIdx.x * blockDim.x + threadIdx.x;
    if (e < E) atomicAdd(&deg[dst[e]], 1.0f);
}

// ---------------------------------------------------------------- deg -> deg^{-1/2} in place
__global__ void k_rsqrt_inplace(float* __restrict__ d, int n) {
    int i = blockIdx.x * blockDim.x + threadIdx.x;
    if (i < n) d[i] = rsqrtf(d[i]);   // deg >= 1 always (self loop)
}

// ---------------------------------------------------------------- per-edge symmetric norm
__global__ void k_norm(const int* __restrict__ src, const int* __restrict__ dst,
                       const float* __restrict__ dinv, float* __restrict__ norm, int E) {
    int e = blockIdx.x * blockDim.x + threadIdx.x;
    if (e < E) norm[e] = dinv[src[e]] * dinv[dst[e]];
}

// ---------------------------------------------------------------- fp32 WMMA GEMM: C[M,256] = A[M,K] @ B[K,256]
// One block (8 waves) per 16-row M tile; wave w computes columns [32w, 32w+32).
// A tile staged in LDS with +4 row pad -> WMMA A-fetch hits all 64 banks once (conflict-free).
template <int K>
__global__ __launch_bounds__(256) void k_gemm_wmma(const float* __restrict__ A,
                                                   const float* __restrict__ B,
                                                   float* __restrict__ C) {
    constexpr int NPAD = K + 4;
    __shared__ float As[16 * NPAD];

    const int tid   = threadIdx.x;
    const int wave  = tid >> 5;
    const int lane  = tid & 31;
    const int mbase = blockIdx.x * 16;           // M = 50000 = 3125*16, no guard needed

    // cooperative float4 load of the 16 x K A tile
    constexpr int NF4 = 16 * K / 4;
    const float4* A4 = reinterpret_cast<const float4*>(A + (size_t)mbase * K);
    #pragma unroll
    for (int j = tid; j < NF4; j += 256) {
        const int row = j / (K / 4);
        const int c4  = j % (K / 4);
        float4 v = A4[(size_t)row * (K / 4) + c4];
        float* d = &As[row * NPAD + c4 * 4];
        d[0] = v.x; d[1] = v.y; d[2] = v.z; d[3] = v.w;
    }
    __syncthreads();

    const int half = lane >> 4;                  // 0: K pair {k,k+1}; 1: K pair {k+2,k+3}
    const int l15  = lane & 15;
    const int n0   = wave * 32;

    v8f acc0 = {};
    v8f acc1 = {};

    #pragma unroll 4
    for (int k0 = 0; k0 < K; k0 += 4) {
        const int kr = k0 + 2 * half;
        // A operand: lane (l15, half) holds A[l15][kr], A[l15][kr+1]
        const float* ap = &As[l15 * NPAD + kr];
        v2f a; a.x = ap[0]; a.y = ap[1];
        // B operand for tile 0 and tile 1: lane holds B[kr][n], B[kr+1][n]
        const float* brow0 = B + (size_t)kr * 256;
        const float* brow1 = brow0 + 256;
        v2f b0, b1;
        b0.x = brow0[n0 + l15];      b0.y = brow1[n0 + l15];
        b1.x = brow0[n0 + 16 + l15]; b1.y = brow1[n0 + 16 + l15];

        acc0 = __builtin_amdgcn_wmma_f32_16x16x4_f32(false, a, false, b0, (short)0, acc0, false, false);
        acc1 = __builtin_amdgcn_wmma_f32_16x16x4_f32(false, a, false, b1, (short)0, acc1, false, false);
    }

    // D layout: VGPR r -> (M = r + 8*half, N = l15)
    #pragma unroll
    for (int r = 0; r < 8; ++r) {
        const size_t m = (size_t)(mbase + r + 8 * half);
        C[m * 256 + n0 + l15]      = acc0[r];
        C[m * 256 + n0 + 16 + l15] = acc1[r];
    }
}

// ---------------------------------------------------------------- edge-parallel scatter-add
// thread -> (edge e, chunk c of 4 floats); 64 consecutive threads share one edge,
// so each wave reads half an h[src] row coalesced and atomics half an out[dst] row.
__global__ void k_aggregate(const float4* __restrict__ h4, const int* __restrict__ src,
                            const int* __restrict__ dst, const float* __restrict__ norm,
                            float* __restrict__ agg, int E) {
    const int gid = blockIdx.x * blockDim.x + threadIdx.x;
    const int e = gid >> 6;
    const int c = gid & 63;
    if (e >= E) return;
    const int   s = src[e];
    const int   d = dst[e];
    const float w = norm[e];
    float4 v = h4[(size_t)s * 64 + c];
    float* o = &agg[(size_t)d * 256 + c * 4];
    atomicAdd(o + 0, v.x * w);
    atomicAdd(o + 1, v.y * w);
    atomicAdd(o + 2, v.z * w);
    atomicAdd(o + 3, v.w * w);
}

// ---------------------------------------------------------------- out = [relu](agg + h*dinv^2 + b)
__global__ void k_combine(const float4* __restrict__ agg4, const float4* __restrict__ h4,
                          const float* __restrict__ dinv, const float4* __restrict__ bias4,
                          float4* __restrict__ out4, int relu) {
    const int gid = blockIdx.x * blockDim.x + threadIdx.x;
    if (gid >= GCN_N_NODES * 64) return;
    const int i = gid >> 6;
    const int c = gid & 63;
    const float di = dinv[i];
    const float sl = di * di;                   // self-loop norm
    float4 a = agg4[gid];
    float4 h = h4[gid];
    float4 b = bias4[c];
    float4 r;
    r.x = a.x + h.x * sl + b.x;
    r.y = a.y + h.y * sl + b.y;
    r.z = a.z + h.z * sl + b.z;
    r.w = a.w + h.w * sl + b.w;
    if (relu) {
        r.x = fmaxf(r.x, 0.0f); r.y = fmaxf(r.y, 0.0f);
        r.z = fmaxf(r.z, 0.0f); r.w = fmaxf(r.w, 0.0f);
    }
    out4[gid] = r;
}

// ---------------------------------------------------------------- host launcher
extern "C" void kernel_launch(void* const* d_in, const int* in_sizes, int n_in,
                              void* d_out, int out_size, void* d_ws, size_t ws_size,
                              hipStream_t stream) {
    const float* x   = (const float*)d_in[0];
    const int*   ei  = (const int*)d_in[1];          // per harness contract: integer -> const int*
    const float* W1  = (const float*)d_in[2];
    const float* b1  = (const float*)d_in[3];
    const float* W2  = (const float*)d_in[4];
    const float* b2  = (const float*)d_in[5];
    float*       out = (float*)d_out;

    const int* src = ei;
    const int* dst = ei + GCN_N_EDGES;

    // workspace layout (floats); d_out is reused as the second ping-pong buffer
    float* ws   = (float*)d_ws;
    float* dinv = ws;                                      // 50176 (aligned region)
    float* norm = dinv + 50176;                            // 800000
    float* bufA = norm + GCN_N_EDGES;                      // 50000*256
    float* bufB = out;                                     // 50000*256 (d_out)

    const int T = 256;
    const int gDeg4   = (GCN_N_NODES / 4 + T - 1) / T;          // deg fill: 12500 float4
    const int gEdges  = (GCN_N_EDGES + T - 1) / T;              // 3125
    const int gNodes  = (GCN_N_NODES + T - 1) / T;              // 196
    const int gGemm   = GCN_N_NODES / 16;                       // 3125
    const int gFeat4  = (GCN_N_NODES * 64) / T;                 // 12500 (exact)
    const int gAgg    = (GCN_N_EDGES * 64) / T;                 // 200000 (exact)

    // degrees (self loop contributes 1.0) -> dinv
    k_fill_f4<<<gDeg4, T, 0, stream>>>((float4*)dinv, 1.0f, GCN_N_NODES / 4);
    k_deg_count<<<gEdges, T, 0, stream>>>(dst, dinv, GCN_N_EDGES);
    k_rsqrt_inplace<<<gNodes, T, 0, stream>>>(dinv, GCN_N_NODES);
    k_norm<<<gEdges, T, 0, stream>>>(src, dst, dinv, norm, GCN_N_EDGES);

    // ---- layer 1: h1 = x @ W1 -> bufA ; agg -> bufB(d_out) ; relu(agg + self + b1) -> bufA
    k_gemm_wmma<GCN_IN_CH><<<gGemm, T, 0, stream>>>(x, W1, bufA);
    k_fill_f4<<<gFeat4, T, 0, stream>>>((float4*)bufB, 0.0f, GCN_N_NODES * 64);
    k_aggregate<<<gAgg, T, 0, stream>>>((const float4*)bufA, src, dst, norm, bufB, GCN_N_EDGES);
    k_combine<<<gFeat4, T, 0, stream>>>((const float4*)bufB, (const float4*)bufA, dinv,
                                        (const float4*)b1, (float4*)bufA, 1);

    // ---- layer 2: h2 = h_relu @ W2 -> bufB(d_out) ; agg -> bufA ; (agg + self + b2) -> d_out
    k_gemm_wmma<GCN_HID><<<gGemm, T, 0, stream>>>(bufA, W2, bufB);
    k_fill_f4<<<gFeat4, T, 0, stream>>>((float4*)bufA, 0.0f, GCN_N_NODES * 64);
    k_aggregate<<<gAgg, T, 0, stream>>>((const float4*)bufB, src, dst, norm, bufA, GCN_N_EDGES);
    k_combine<<<gFeat4, T, 0, stream>>>((const float4*)bufA, (const float4*)bufB, dinv,
                                        (const float4*)b2, (float4*)out, 0);
}